// FIFOMemory_60790967107884
// MI455X (gfx1250) — compile-verified
//
#include <hip/hip_runtime.h>

// ---------------- problem constants (match reference) ----------------
#define MEM_N   131072
#define QN      2048
#define DIM     64
#define BATCH   4096
#define TOPK    10
#define NSLICE  32
#define SLICE_LEN (MEM_N / NSLICE)        // 4096 rows per slice
#define QTILE   16
#define WAVES   8
#define ROWS_PER_ITER (WAVES * 16)        // 128 memory rows per block-iteration
#define ITERS   (SLICE_LEN / ROWS_PER_ITER) // 32

typedef __attribute__((ext_vector_type(2))) float v2f;
typedef __attribute__((ext_vector_type(8))) float v8f;

#define NEG_INF (-__builtin_inff())

// Virtual FIFO write: row n comes from embeddings if (n - wp) mod N < BATCH.
__device__ __forceinline__ const float* fifo_row(const float* __restrict__ memory,
                                                 const float* __restrict__ emb,
                                                 int wp, int n) {
  int diff = n - wp;
  if (diff < 0) diff += MEM_N;
  return (diff < BATCH) ? (emb + (size_t)diff * DIM) : (memory + (size_t)n * DIM);
}

// Descending top-k shift-insert; arrays indexed only with compile-time
// constants so they live in VGPRs. Tie-break: smaller index wins.
__device__ __forceinline__ void topk_insert(float (&vals)[TOPK], int (&idxs)[TOPK],
                                            float v, int gi) {
  // fast reject against current minimum
  bool beats_last = (v > vals[TOPK - 1]) ||
                    (v == vals[TOPK - 1] && gi < idxs[TOPK - 1]);
  if (!beats_last) return;
  float cv = v; int ci = gi;
#pragma unroll
  for (int p = 0; p < TOPK; ++p) {
    bool better = (cv > vals[p]) || (cv == vals[p] && ci < idxs[p]);
    if (better) {
      float tvv = vals[p]; int tii = idxs[p];
      vals[p] = cv; idxs[p] = ci;
      cv = tvv; ci = tii;
    }
  }
}

// ---------------- kernel 1: inverse L2 norms (one wave32 per row) ----------------
__global__ void fifo_norms_kernel(const float* __restrict__ memory,
                                  const float* __restrict__ emb,
                                  const float* __restrict__ query,
                                  const int* __restrict__ wp_p,
                                  float* __restrict__ inv_mnorm,
                                  float* __restrict__ inv_qnorm) {
  const int wp   = *wp_p;
  const int lane = threadIdx.x & 31;
  const int wave = threadIdx.x >> 5;
  const int row  = blockIdx.x * WAVES + wave;
  const int total = MEM_N + QN;
  if (row >= total) return;

  const float* src = (row < MEM_N) ? fifo_row(memory, emb, wp, row)
                                   : (query + (size_t)(row - MEM_N) * DIM);
  v2f v = *(const v2f*)(src + lane * 2);
  float s = v.x * v.x + v.y * v.y;
#pragma unroll
  for (int m = 16; m >= 1; m >>= 1) s += __shfl_xor(s, m, 32);
  if (lane == 0) {
    float inv = 1.0f / fmaxf(sqrtf(s), 1e-12f);
    if (row < MEM_N) inv_mnorm[row] = inv;
    else             inv_qnorm[row - MEM_N] = inv;
  }
}

// ---------------- kernel 2: fused WMMA cosine scores + per-slice top-k ----------------
// grid: (QN/QTILE, NSLICE), block: 256 = 8 waves.
// Each wave computes 16x16 score tiles with 16 chained V_WMMA_F32_16X16X4_F32.
__global__ void __launch_bounds__(256)
fifo_scores_topk_kernel(const float* __restrict__ memory,
                        const float* __restrict__ emb,
                        const float* __restrict__ query,
                        const int* __restrict__ wp_p,
                        const float* __restrict__ inv_mnorm,
                        const float* __restrict__ inv_qnorm,
                        float* __restrict__ candVal,
                        int* __restrict__ candIdx) {
  __shared__ float scoreBuf[WAVES * QTILE * 16];   //  8 KB
  __shared__ float mergeVal[QTILE * 16 * TOPK];    // 10 KB
  __shared__ int   mergeIdx[QTILE * 16 * TOPK];    // 10 KB

  const int wp     = *wp_p;
  const int tid    = threadIdx.x;
  const int lane   = tid & 31;
  const int wave   = tid >> 5;
  const int half   = (lane >> 4) & 1;   // 0: lanes 0-15, 1: lanes 16-31
  const int lane16 = lane & 15;
  const int qtile  = blockIdx.x;
  const int slice  = blockIdx.y;
  const int sliceBase = slice * SLICE_LEN;

  // A tile: 16 normalized query rows. ISA 32-bit A 16x4 layout:
  // lanes 0-15 hold K={0,1} (vgpr0,1); lanes 16-31 hold K={2,3}.
  // => per wmma chunk kc, lane supplies elements [4*kc + 2*half + {0,1}].
  float aA[32];
  {
    const int   qg = qtile * QTILE + lane16;
    const float qs = inv_qnorm[qg];
    const v2f*  qrow = (const v2f*)(query + (size_t)qg * DIM);
#pragma unroll
    for (int kc = 0; kc < 16; ++kc) {
      v2f v = qrow[2 * kc + half];
      aA[2 * kc]     = v.x * qs;
      aA[2 * kc + 1] = v.y * qs;
    }
  }

  // Per-thread partial top-k. Thread (sq, sj) owns query sq, tile-column sj.
  float tv[TOPK]; int ti[TOPK];
#pragma unroll
  for (int p = 0; p < TOPK; ++p) { tv[p] = NEG_INF; ti[p] = 0x7FFFFFFF; }
  const int sq = tid >> 4;
  const int sj = tid & 15;

  for (int it = 0; it < ITERS; ++it) {
    const int nBase = sliceBase + it * ROWS_PER_ITER;

    // B tile: 16 normalized memory rows for this wave (B[k][n] = mem_norm[row n][k]).
    const int   n   = nBase + wave * 16 + lane16;
    const float ms  = inv_mnorm[n];
    const v2f* mrow = (const v2f*)fifo_row(memory, emb, wp, n);
    float bB[32];
#pragma unroll
    for (int kc = 0; kc < 16; ++kc) {
      v2f v = mrow[2 * kc + half];
      bB[2 * kc]     = v.x * ms;
      bB[2 * kc + 1] = v.y * ms;
    }

    // 16 chained f32 WMMAs accumulate the full K=64 dot products.
    v8f c = {0.f, 0.f, 0.f, 0.f, 0.f, 0.f, 0.f, 0.f};
#pragma unroll
    for (int kc = 0; kc < 16; ++kc) {
      v2f a; a.x = aA[2 * kc]; a.y = aA[2 * kc + 1];
      v2f b; b.x = bB[2 * kc]; b.y = bB[2 * kc + 1];
      c = __builtin_amdgcn_wmma_f32_16x16x4_f32(
              /*neg_a=*/false, a, /*neg_b=*/false, b,
              /*c_mod=*/(short)0, c, /*reuse_a=*/false, /*reuse_b=*/false);
    }

    // C/D layout: vgpr r, lanes 0-15 -> M=r, N=lane; lanes 16-31 -> M=r+8.
#pragma unroll
    for (int r = 0; r < 8; ++r)
      scoreBuf[(wave * QTILE + (r + 8 * half)) * 16 + lane16] = c[r];
    __syncthreads();

#pragma unroll
    for (int w = 0; w < WAVES; ++w) {
      float v  = scoreBuf[(w * QTILE + sq) * 16 + sj];
      int   gi = nBase + w * 16 + sj;
      topk_insert(tv, ti, v, gi);
    }
    __syncthreads();
  }

  // Merge 16 column-owner partials per query -> per-slice top-k candidates.
#pragma unroll
  for (int p = 0; p < TOPK; ++p) {
    mergeVal[(sq * 16 + sj) * TOPK + p] = tv[p];
    mergeIdx[(sq * 16 + sj) * TOPK + p] = ti[p];
  }
  __syncthreads();

  if (tid < QTILE) {
    float fv[TOPK]; int fi[TOPK];
#pragma unroll
    for (int p = 0; p < TOPK; ++p) { fv[p] = NEG_INF; fi[p] = 0x7FFFFFFF; }
    for (int j = 0; j < 16; ++j)
      for (int p = 0; p < TOPK; ++p)
        topk_insert(fv, fi, mergeVal[(tid * 16 + j) * TOPK + p],
                            mergeIdx[(tid * 16 + j) * TOPK + p]);
    const int qg = qtile * QTILE + tid;
    float* cv = candVal + ((size_t)qg * NSLICE + slice) * TOPK;
    int*   ci = candIdx + ((size_t)qg * NSLICE + slice) * TOPK;
#pragma unroll
    for (int p = 0; p < TOPK; ++p) { cv[p] = fv[p]; ci[p] = fi[p]; }
  }
}

// ---------------- kernel 3: final merge + gather of raw FIFO rows ----------------
// grid: QN blocks, block: DIM(=64) threads.
__global__ void fifo_merge_gather_kernel(const float* __restrict__ memory,
                                         const float* __restrict__ emb,
                                         const int* __restrict__ wp_p,
                                         const float* __restrict__ candVal,
                                         const int* __restrict__ candIdx,
                                         float* __restrict__ outRetr,
                                         float* __restrict__ outVals) {
  __shared__ float sVal[TOPK];
  __shared__ int   sIdx[TOPK];
  const int q   = blockIdx.x;
  const int tid = threadIdx.x;

  if (tid == 0) {
    float fv[TOPK]; int fi[TOPK];
#pragma unroll
    for (int p = 0; p < TOPK; ++p) { fv[p] = NEG_INF; fi[p] = 0x7FFFFFFF; }
    const float* cv = candVal + (size_t)q * NSLICE * TOPK;
    const int*   ci = candIdx + (size_t)q * NSLICE * TOPK;
    for (int e = 0; e < NSLICE * TOPK; ++e) topk_insert(fv, fi, cv[e], ci[e]);
#pragma unroll
    for (int p = 0; p < TOPK; ++p) { sVal[p] = fv[p]; sIdx[p] = fi[p]; }
  }
  __syncthreads();

  const int wp = *wp_p;
  for (int t = 0; t < TOPK; ++t) {
    const float* src = fifo_row(memory, emb, wp, sIdx[t]);
    outRetr[((size_t)q * TOPK + t) * DIM + tid] = src[tid];
  }
  if (tid < TOPK) outVals[(size_t)q * TOPK + tid] = sVal[tid];
}

// ---------------- launch ----------------
extern "C" void kernel_launch(void* const* d_in, const int* in_sizes, int n_in,
                              void* d_out, int out_size, void* d_ws, size_t ws_size,
                              hipStream_t stream) {
  const float* memory = (const float*)d_in[0];
  const float* emb    = (const float*)d_in[1];
  const float* query  = (const float*)d_in[2];
  const int*   wp     = (const int*)d_in[3];
  // d_in[4] = top_k (10) -> fixed in output layout, unused at runtime.

  float* out      = (float*)d_out;
  float* outRetr  = out;                                    // [QN, TOPK, DIM]
  float* outVals  = out + (size_t)QN * TOPK * DIM;          // [QN, TOPK]

  // Workspace layout (~5.6 MB total).
  float* inv_mnorm = (float*)d_ws;                          // MEM_N floats
  float* inv_qnorm = inv_mnorm + MEM_N;                     // QN floats
  float* candVal   = inv_qnorm + QN;                        // QN*NSLICE*TOPK floats
  int*   candIdx   = (int*)(candVal + (size_t)QN * NSLICE * TOPK);

  dim3 g1((MEM_N + QN) / WAVES);                            // 16640 blocks
  fifo_norms_kernel<<<g1, 256, 0, stream>>>(memory, emb, query, wp,
                                            inv_mnorm, inv_qnorm);

  dim3 g2(QN / QTILE, NSLICE);                              // 128 x 32 blocks
  fifo_scores_topk_kernel<<<g2, 256, 0, stream>>>(memory, emb, query, wp,
                                                  inv_mnorm, inv_qnorm,
                                                  candVal, candIdx);

  fifo_merge_gather_kernel<<<QN, DIM, 0, stream>>>(memory, emb, wp,
                                                   candVal, candIdx,
                                                   outRetr, outVals);
}